// yololoss_66477503807536
// MI455X (gfx1250) — compile-verified
//
#include <hip/hip_runtime.h>
#include <stdint.h>

// YOLOv1 loss, memory-bound streaming reduction.
// Kernel 1: each block stages 256 cells (2 x 30 KB) into LDS via
//           global_load_async_to_lds_b64 (wave-self-contained, so only
//           s_wait_asynccnt is needed), computes per-cell loss, reduces the
//           wave with V_WMMA_F32_16X16X4_F32 (B = ones), writes 1 float/block.
// Kernel 2: single wave reduces block partials (same WMMA reducer), scales by
//           1/N, writes the scalar. Fully deterministic (no float atomics).

typedef float v2f __attribute__((ext_vector_type(2)));
typedef float v8f __attribute__((ext_vector_type(8)));

// Exact f32 sum of one value per lane across the wave, using WMMA with an
// all-ones B matrix:  D[m][n] = sum_k A[m][k]  (independent of n and of the
// A-operand layout).  Per-lane sum of the 8 C VGPRs gives half a column sum;
// pairing lane l with lane l^16 completes the column = total sum.
__device__ __forceinline__ float wave_sum_wmma(float s) {
  v2f a; a[0] = s;    a[1] = 0.0f;
  v2f b; b[0] = 1.0f; b[1] = 1.0f;
  v8f c = {};
  c = __builtin_amdgcn_wmma_f32_16x16x4_f32(false, a, false, b,
                                            (short)0, c, false, false);
  float u = ((c[0] + c[1]) + (c[2] + c[3])) + ((c[4] + c[5]) + (c[6] + c[7]));
  u += __shfl_xor(u, 16, 32);
  return u;   // every lane holds the full wave sum
}

#define NCH 30          // channels per cell
#define CPB 256         // cells per block (== blockDim.x)
#define WAVE_BYTES (32 * NCH * 4)   // 3840 B staged per wave per tensor

__global__ void yolo_loss_partial(const float* __restrict__ pred,
                                  const float* __restrict__ target,
                                  float* __restrict__ partial,
                                  int ncells) {
  __shared__ __align__(16) float sp[CPB * NCH];   // 30720 B
  __shared__ __align__(16) float st[CPB * NCH];   // 30720 B
  __shared__ float wsum[CPB / 32];

  const int tid  = threadIdx.x;
  const int lane = tid & 31;
  const int wave = tid >> 5;

  const long long blockCell = (long long)blockIdx.x * CPB;
  const size_t totalB = (size_t)ncells * (NCH * 4);
  const size_t gbase  = (size_t)blockCell * (NCH * 4) + (size_t)wave * WAVE_BYTES;

  const unsigned sp_lds = (unsigned)(uintptr_t)sp + (unsigned)wave * WAVE_BYTES;
  const unsigned st_lds = (unsigned)(uintptr_t)st + (unsigned)wave * WAVE_BYTES;

  // --- async stage: 15 coalesced b64 loads per lane per tensor -------------
#pragma unroll
  for (int k = 0; k < WAVE_BYTES / (32 * 8); ++k) {   // 15 iterations
    const unsigned off = (unsigned)(k * 32 + lane) * 8u;
    if (gbase + off + 8 <= totalB) {
      const unsigned long long gp =
          (unsigned long long)(uintptr_t)pred + gbase + off;
      const unsigned long long gt =
          (unsigned long long)(uintptr_t)target + gbase + off;
      asm volatile("global_load_async_to_lds_b64 %0, %1, off"
                   :: "v"(sp_lds + off), "v"(gp) : "memory");
      asm volatile("global_load_async_to_lds_b64 %0, %1, off"
                   :: "v"(st_lds + off), "v"(gt) : "memory");
    }
  }
  asm volatile("s_wait_asynccnt 0" ::: "memory");
  // each wave consumes only the LDS it loaded itself -> no barrier needed

  // --- per-cell loss -------------------------------------------------------
  float loss = 0.0f;
  const long long cell = blockCell + tid;
  if (cell < (long long)ncells) {
    const float* p = sp + tid * NCH;
    const float* t = st + tid * NCH;

    const float conf = t[4];          // exactly 0.0 or 1.0 by construction
    if (conf == 0.0f) {
      const float d4 = p[4] - t[4];
      const float d9 = p[9] - t[9];
      loss = 0.5f * (d4 * d4 + d9 * d9);
    } else {
      // class loss (channels 10..29)
      float lc = 0.0f;
#pragma unroll
      for (int c = 10; c < 30; ++c) {
        const float d = t[c] - p[c];
        lc += d * d;
      }
      // IoU of both predicted boxes vs target box 0 (unclamped intersection)
      const float t_ltx = t[0] - t[2] * 0.5f, t_lty = t[1] - t[3] * 0.5f;
      const float t_rbx = t[0] + t[2] * 0.5f, t_rby = t[1] + t[3] * 0.5f;
      const float area_t = t[2] * t[3];
      float iou[2];
#pragma unroll
      for (int bb = 0; bb < 2; ++bb) {
        const float* pb = p + 5 * bb;
        const float ltx = fmaxf(pb[0] - pb[2] * 0.5f, t_ltx);
        const float lty = fmaxf(pb[1] - pb[3] * 0.5f, t_lty);
        const float rbx = fminf(pb[0] + pb[2] * 0.5f, t_rbx);
        const float rby = fminf(pb[1] + pb[3] * 0.5f, t_rby);
        const float inter = (rbx - ltx) * (rby - lty);
        const float area_p = pb[2] * pb[3];
        iou[bb] = inter / (area_p + area_t - inter);
      }
      const int r  = (iou[1] > iou[0]) ? 1 : 0;   // argmax, tie -> box 0
      const int nr = 1 - r;

      const float* pr = p + 5 * r;
      const float* tr = t + 5 * r;
      const float dx = tr[0] - pr[0];
      const float dy = tr[1] - pr[1];
      const float dw = sqrtf(tr[2]) - sqrtf(pr[2]);
      const float dh = sqrtf(tr[3]) - sqrtf(pr[3]);
      const float dc = tr[4] - pr[4];
      const float dcn = t[5 * nr + 4] - p[5 * nr + 4];

      loss = lc + 5.0f * (dx * dx + dy * dy + dw * dw + dh * dh)
                + dc * dc + 0.5f * dcn * dcn;
    }
  }

  // --- wave reduce (WMMA) + block reduce ----------------------------------
  const float ws = wave_sum_wmma(loss);   // EXEC is full here (no early exit)
  if (lane == 0) wsum[wave] = ws;
  __syncthreads();
  if (tid == 0) {
    float s = 0.0f;
#pragma unroll
    for (int w = 0; w < CPB / 32; ++w) s += wsum[w];
    partial[blockIdx.x] = s;
  }
}

__global__ void yolo_loss_final(const float* __restrict__ partial, int n,
                                float scale, float* __restrict__ out) {
  const int lane = threadIdx.x;   // launched with exactly 32 threads
  float s = 0.0f;
  for (int i = lane; i < n; i += 32) s += partial[i];
  const float total = wave_sum_wmma(s);
  if (lane == 0) out[0] = total * scale;
}

extern "C" void kernel_launch(void* const* d_in, const int* in_sizes, int n_in,
                              void* d_out, int out_size, void* d_ws, size_t ws_size,
                              hipStream_t stream) {
  (void)n_in; (void)out_size; (void)ws_size;
  const float* pred   = (const float*)d_in[0];
  const float* target = (const float*)d_in[1];
  float* out = (float*)d_out;
  float* partial = (float*)d_ws;

  const int total  = in_sizes[0];        // N*H*W*C
  const int ncells = total / NCH;        // N*H*W
  const int nblk   = (ncells + CPB - 1) / CPB;
  const int nbatch = ncells / (14 * 14); // N
  const float scale = 1.0f / (float)nbatch;

  yolo_loss_partial<<<nblk, CPB, 0, stream>>>(pred, target, partial, ncells);
  yolo_loss_final<<<1, 32, 0, stream>>>(partial, nblk, scale, out);
}